// GNN_node_57037165691354
// MI455X (gfx1250) — compile-verified
//
#include <hip/hip_runtime.h>
#include <hip/hip_bf16.h>

typedef __attribute__((ext_vector_type(2))) float v2f;
typedef __attribute__((ext_vector_type(8))) float v8f;

#define WMMA_F32(a, b, c) \
    __builtin_amdgcn_wmma_f32_16x16x4_f32(false, (a), false, (b), (short)0, (c), false, false)

// ---------------------------------------------------------------------------
// WMMA fp32 GEMM:  C[Nrows,M] = op(A)[Nrows,K] @ W[K,M] + bias
// op(A) optionally applies per-K affine + ReLU (fused BN of previous stage).
// Each wave computes a 32x64 strip = 8 independent 16x16 WMMA tiles:
//   - 8 accumulator chains (full XDL latency hiding)
//   - each B fragment reused by 2 A tiles (B traffic per FLOP halved)
//   - software-pipelined A/prologue prefetch (next step's A issued before
//     this step's WMMAs) so A latency is covered by the WMMA group
//   - B via two running pointers with immediate offsets, no per-iter mul
// Block = 256 threads = 8 waves.  K, M compile-time (128/256 here).
// ---------------------------------------------------------------------------
template <int K, int M, bool PRO>
__global__ __launch_bounds__(256) void gemm_wmma_f32(
    const float* __restrict__ A, const float* __restrict__ W,
    const float* __restrict__ bias, float* __restrict__ C,
    int Nrows, const float* __restrict__ pro)
{
    constexpr int colGroups = M >> 6;               // 64-col groups (2 or 4)
    constexpr int rowSlotsPerBlock = 8 / colGroups; // 32-row slots per block
    const int lane = threadIdx.x & 31;
    const int wave = threadIdx.x >> 5;
    const int cg = wave % colGroups;
    const int rs = wave / colGroups;
    const int row0 = (blockIdx.x * rowSlotsPerBlock + rs) << 5;
    if (row0 >= Nrows) return;                      // wave-uniform guard
    const bool store2 = (row0 + 16) < Nrows;        // second 16-row tile valid?
    const int row1 = store2 ? (row0 + 16) : row0;   // clamp: stay in bounds
    const int col0 = cg << 6;
    const int r    = lane & 15;                     // A row / B col in tile
    const int kr   = (lane >> 4) << 1;              // 0 or 2 (K sub-offset)

    const float* __restrict__ a0p = A + (size_t)(row0 + r) * K + kr;
    const float* __restrict__ a1p = A + (size_t)(row1 + r) * K + kr;
    const float* __restrict__ bp0 = W + (size_t)kr * M + col0 + r;
    const float* __restrict__ bp1 = bp0 + M;
    const float* __restrict__ pp  = PRO ? (pro + kr) : nullptr;
    constexpr long bstep = 4L * M;

    v8f acc00 = {}, acc01 = {}, acc02 = {}, acc03 = {};
    v8f acc10 = {}, acc11 = {}, acc12 = {}, acc13 = {};

    // --- software pipeline: preload first A (+ prologue params) ------------
    v2f ra0 = *(const v2f*)a0p;  a0p += 4;
    v2f ra1 = *(const v2f*)a1p;  a1p += 4;
    v2f rsc, rsh;
    if (PRO) { rsc.x = pp[0]; rsc.y = pp[1]; rsh.x = pp[K]; rsh.y = pp[K + 1]; pp += 4; }

    #pragma unroll 4
    for (int k = 0; k < K - 4; k += 4) {
        v2f b0, b1, b2, b3;
        b0.x = bp0[0];  b0.y = bp1[0];
        b1.x = bp0[16]; b1.y = bp1[16];
        b2.x = bp0[32]; b2.y = bp1[32];
        b3.x = bp0[48]; b3.y = bp1[48];
        // prefetch next step's A (+ prologue) before this step's WMMAs
        v2f na0 = *(const v2f*)a0p;
        v2f na1 = *(const v2f*)a1p;
        v2f nsc, nsh;
        if (PRO) { nsc.x = pp[0]; nsc.y = pp[1]; nsh.x = pp[K]; nsh.y = pp[K + 1]; }

        v2f a0 = ra0, a1 = ra1;
        if (PRO) {
            a0.x = fmaxf(ra0.x * rsc.x + rsh.x, 0.0f);
            a0.y = fmaxf(ra0.y * rsc.y + rsh.y, 0.0f);
            a1.x = fmaxf(ra1.x * rsc.x + rsh.x, 0.0f);
            a1.y = fmaxf(ra1.y * rsc.y + rsh.y, 0.0f);
        }
        acc00 = WMMA_F32(a0, b0, acc00);
        acc10 = WMMA_F32(a1, b0, acc10);
        acc01 = WMMA_F32(a0, b1, acc01);
        acc11 = WMMA_F32(a1, b1, acc11);
        acc02 = WMMA_F32(a0, b2, acc02);
        acc12 = WMMA_F32(a1, b2, acc12);
        acc03 = WMMA_F32(a0, b3, acc03);
        acc13 = WMMA_F32(a1, b3, acc13);

        ra0 = na0; ra1 = na1;
        if (PRO) { rsc = nsc; rsh = nsh; pp += 4; }
        a0p += 4; a1p += 4; bp0 += bstep; bp1 += bstep;
    }
    // --- tail step (k = K-4): no further prefetch --------------------------
    {
        v2f b0, b1, b2, b3;
        b0.x = bp0[0];  b0.y = bp1[0];
        b1.x = bp0[16]; b1.y = bp1[16];
        b2.x = bp0[32]; b2.y = bp1[32];
        b3.x = bp0[48]; b3.y = bp1[48];
        v2f a0 = ra0, a1 = ra1;
        if (PRO) {
            a0.x = fmaxf(ra0.x * rsc.x + rsh.x, 0.0f);
            a0.y = fmaxf(ra0.y * rsc.y + rsh.y, 0.0f);
            a1.x = fmaxf(ra1.x * rsc.x + rsh.x, 0.0f);
            a1.y = fmaxf(ra1.y * rsc.y + rsh.y, 0.0f);
        }
        acc00 = WMMA_F32(a0, b0, acc00);
        acc10 = WMMA_F32(a1, b0, acc10);
        acc01 = WMMA_F32(a0, b1, acc01);
        acc11 = WMMA_F32(a1, b1, acc11);
        acc02 = WMMA_F32(a0, b2, acc02);
        acc12 = WMMA_F32(a1, b2, acc12);
        acc03 = WMMA_F32(a0, b3, acc03);
        acc13 = WMMA_F32(a1, b3, acc13);
    }

    const float bc0 = bias[col0 + r];
    const float bc1 = bias[col0 + 16 + r];
    const float bc2 = bias[col0 + 32 + r];
    const float bc3 = bias[col0 + 48 + r];
    const int rowAdd = (lane >> 4) << 3;            // lanes 16-31 hold rows +8
    #pragma unroll
    for (int i = 0; i < 8; ++i) {
        float* __restrict__ cp = C + (size_t)(row0 + i + rowAdd) * M + col0 + r;
        cp[0]  = acc00[i] + bc0;
        cp[16] = acc01[i] + bc1;
        cp[32] = acc02[i] + bc2;
        cp[48] = acc03[i] + bc3;
    }
    if (store2) {
        #pragma unroll
        for (int i = 0; i < 8; ++i) {
            float* __restrict__ cp = C + (size_t)(row1 + i + rowAdd) * M + col0 + r;
            cp[0]  = acc10[i] + bc0;
            cp[16] = acc11[i] + bc1;
            cp[32] = acc12[i] + bc2;
            cp[48] = acc13[i] + bc3;
        }
    }
}

// z = (1 + eps[l]) * h   (vectorized float4)
__global__ void init_z_kernel(const float* __restrict__ h, float* __restrict__ z,
                              const float* __restrict__ eps, int l, int total4)
{
    int i = blockIdx.x * blockDim.x + threadIdx.x;
    if (i >= total4) return;
    const float s = 1.0f + eps[l];
    float4 v = ((const float4*)h)[i];
    v.x *= s; v.y *= s; v.z *= s; v.w *= s;
    ((float4*)z)[i] = v;
}

// One wave per edge: gather float4 of h[src], ReLU, scatter-add into z[dst].
__global__ void scatter_relu_add(const float* __restrict__ h,
                                 const long long* __restrict__ ei,
                                 int E, float* __restrict__ z)
{
    const long long t = (long long)blockIdx.x * blockDim.x + threadIdx.x;
    const long long e = t >> 5;
    if (e >= E) return;
    const int f = (int)(t & 31) << 2;               // feature offset 0..124
    const long long s = ei[e];
    const long long d = ei[(long long)E + e];
    const float4 v = *(const float4*)(h + (size_t)s * 128 + f);
    float* __restrict__ out = z + (size_t)d * 128 + f;
    atomicAdd(out + 0, fmaxf(v.x, 0.0f));
    atomicAdd(out + 1, fmaxf(v.y, 0.0f));
    atomicAdd(out + 2, fmaxf(v.z, 0.0f));
    atomicAdd(out + 3, fmaxf(v.w, 0.0f));
}

__global__ void zero_kernel(float* __restrict__ p, int count)
{
    int i = blockIdx.x * blockDim.x + threadIdx.x;
    if (i < count) p[i] = 0.0f;
}

// Column-wise sum / sumsq; blockDim.x == m (feature count), rows strided by grid.
__global__ void bn_stats(const float* __restrict__ t, int n, int m,
                         float* __restrict__ stats)
{
    const int c = threadIdx.x;
    float s = 0.0f, sq = 0.0f;
    for (int r = blockIdx.x; r < n; r += gridDim.x) {
        const float v = t[(size_t)r * m + c];
        s += v; sq += v * v;
    }
    atomicAdd(&stats[c],     s);
    atomicAdd(&stats[m + c], sq);
}

// Produce fused affine: y = x*ab[c] + ab[m+c]  ==  (x-mean)*rstd*gamma + beta
__global__ void bn_finalize(const float* __restrict__ stats,
                            const float* __restrict__ gamma,
                            const float* __restrict__ beta,
                            int m, float n, float* __restrict__ ab)
{
    int c = blockIdx.x * blockDim.x + threadIdx.x;
    if (c >= m) return;
    const float mean = stats[c] / n;
    const float var  = stats[m + c] / n - mean * mean;
    const float rstd = rsqrtf(var + 1e-5f);
    const float a    = gamma[c] * rstd;
    ab[c]     = a;
    ab[m + c] = beta[c] - mean * a;
}

// out = bn(v) with optional ReLU (m must be power of two; m == 128 here)
__global__ void bn_apply(const float* __restrict__ v, const float* __restrict__ ab,
                         int m, int total, float* __restrict__ out, int do_relu)
{
    int i = blockIdx.x * blockDim.x + threadIdx.x;
    if (i >= total) return;
    const int c = i & (m - 1);
    float y = v[i] * ab[c] + ab[m + c];
    if (do_relu) y = fmaxf(y, 0.0f);
    out[i] = y;
}

static inline int gemm_grid(int nrows, int M)
{
    const int rowSlots = (nrows + 31) / 32;          // 32-row slots
    const int rowSlotsPerBlock = 8 / (M >> 6);
    return (rowSlots + rowSlotsPerBlock - 1) / rowSlotsPerBlock;
}

extern "C" void kernel_launch(void* const* d_in, const int* in_sizes, int n_in,
                              void* d_out, int out_size, void* d_ws, size_t ws_size,
                              hipStream_t stream)
{
    const float*     x    = (const float*)d_in[0];
    const long long* ei   = (const long long*)d_in[1];   // int64 [2, E]
    const float*     fcW  = (const float*)d_in[2];
    const float*     fcb  = (const float*)d_in[3];
    const float*     W1   = (const float*)d_in[4];       // [L,128,256]
    const float*     b1   = (const float*)d_in[5];       // [L,256]
    const float*     g1   = (const float*)d_in[6];
    const float*     bt1  = (const float*)d_in[7];
    const float*     W2   = (const float*)d_in[8];       // [L,256,128]
    const float*     b2   = (const float*)d_in[9];       // [L,128]
    const float*     eps  = (const float*)d_in[10];      // [L]
    const float*     bn_g = (const float*)d_in[11];
    const float*     bn_b = (const float*)d_in[12];
    float*           out  = (float*)d_out;

    const int EMB = 128;
    const int N   = in_sizes[0] / EMB;        // 50000
    const int E   = in_sizes[1] / 2;          // 800000
    const int L   = in_sizes[10];             // 5

    // Workspace carve-up (bytes): B0=h, B1=z/v, B2=t, stats, ab
    const size_t featBytes = (size_t)N * EMB * sizeof(float);   // 25.6 MB
    char* ws = (char*)d_ws;
    float* B0    = (float*)(ws);                                // h      [N,128]
    float* B1    = (float*)(ws + featBytes);                    // z / v  [N,128]
    float* B2    = (float*)(ws + 2 * featBytes);                // t      [N,256]
    float* stats = (float*)(ws + 4 * featBytes);                // 2*256 f32
    float* ab    = (float*)(ws + 4 * featBytes + 2048);         // 2*256 f32

    const int total  = N * EMB;          // 6.4M
    const int total4 = total / 4;
    const dim3 blk(256);

    // fc: h = x @ fcW + fcb
    gemm_wmma_f32<128, 128, false><<<gemm_grid(N, 128), blk, 0, stream>>>(
        x, fcW, fcb, B0, N, nullptr);

    for (int l = 0; l < L; ++l) {
        // z = (1+eps)*h ; scatter-add relu(h[src]) into z[dst]
        init_z_kernel<<<(total4 + 255) / 256, blk, 0, stream>>>(B0, B1, eps, l, total4);
        {
            long long nthreads = (long long)E * 32;
            int nblocks = (int)((nthreads + 255) / 256);
            scatter_relu_add<<<nblocks, blk, 0, stream>>>(B0, ei, E, B1);
        }

        // t = z @ W1[l] + b1[l]            [N,256]
        gemm_wmma_f32<128, 256, false><<<gemm_grid(N, 256), blk, 0, stream>>>(
            B1, W1 + (size_t)l * 128 * 256, b1 + (size_t)l * 256, B2, N, nullptr);

        // BN1 stats -> fused (scale,shift) for the GEMM2 prologue
        zero_kernel<<<2, blk, 0, stream>>>(stats, 512);
        bn_stats<<<512, 256, 0, stream>>>(B2, N, 256, stats);
        bn_finalize<<<1, 256, 0, stream>>>(
            stats, g1 + (size_t)l * 256, bt1 + (size_t)l * 256, 256, (float)N, ab);

        // v = relu(bn1(t)) @ W2[l] + b2[l]  [N,128]  (BN+ReLU fused as prologue)
        gemm_wmma_f32<256, 128, true><<<gemm_grid(N, 128), blk, 0, stream>>>(
            B2, W2 + (size_t)l * 256 * 128, b2 + (size_t)l * 128, B1, N, ab);

        // BN2 (+ReLU except last layer) -> h (or d_out on last layer)
        zero_kernel<<<1, blk, 0, stream>>>(stats, 256);
        bn_stats<<<512, 128, 0, stream>>>(B1, N, 128, stats);
        bn_finalize<<<1, 128, 0, stream>>>(
            stats, bn_g + (size_t)l * 128, bn_b + (size_t)l * 128, 128, (float)N, ab);
        float* dst = (l == L - 1) ? out : B0;
        bn_apply<<<(total + 255) / 256, blk, 0, stream>>>(
            B1, ab, 128, total, dst, (l < L - 1) ? 1 : 0);
    }
}